// ContentSynchronousTimbreFuser_2113123909794
// MI455X (gfx1250) — compile-verified
//
#include <hip/hip_runtime.h>
#include <math.h>

// ---------------------------------------------------------------------------
// ContentSynchronousTimbreFuser for MI455X (gfx1250).
// f32 WMMA GEMM pipeline with async global->LDS tile staging (double-buffered,
// last tile peeled so the steady-state loop is branch-free).
// B=8, S=2048, M=2048, H=512, GH=512. All tensors f32.
// ---------------------------------------------------------------------------

typedef __attribute__((ext_vector_type(2))) float v2f;
typedef __attribute__((ext_vector_type(8))) float v8f;

constexpr int cB  = 8;
constexpr int cS  = 2048;
constexpr int cM  = 2048;
constexpr int cH  = 512;
constexpr int cGH = 512;

constexpr float MIX_BIAS       = -0.25f;
constexpr float VARIATION_BIAS = -1.0f;
constexpr float PRIOR_SCALE    = 1.0f;
constexpr float EPS            = 1e-6f;

// GEMM tiling: 256 threads = 8 waves, block tile 128x64, K block 32.
// Each wave computes a 32x32 patch = 2x2 WMMA 16x16 tiles.
constexpr int TM = 128;
constexpr int TN = 64;
constexpr int TK = 32;
constexpr int A_STRIDE  = 36;  // mult of 4 (16B-aligned async b128 stores), conflict-free b64 frags
constexpr int B_STRIDE  = 72;  // [k][n] layout; 2*72 mod 64 = 16 -> lane halves hit disjoint banks
constexpr int BT_STRIDE = 36;  // [n][k] layout for transposed-B; single b64 frag loads

// low 32 bits of a flat shared-aperture address == LDS byte offset (ISA 10.2)
__device__ __forceinline__ unsigned lds_off(const void* p) {
    return (unsigned)(unsigned long long)p;
}

// CDNA5 async copy: 16B global -> LDS, tracked by ASYNCcnt.
__device__ __forceinline__ void async_g2l_b128(unsigned ldsByteOff, const float* g) {
    asm volatile("global_load_async_to_lds_b128 %0, %1, off"
                 :: "v"(ldsByteOff), "v"(g) : "memory");
}
__device__ __forceinline__ void wait_async_6() {
    asm volatile("s_wait_asynccnt 0x6" ::: "memory");
}
__device__ __forceinline__ void wait_async_0() {
    asm volatile("s_wait_asynccnt 0x0" ::: "memory");
}

// ---------------------------------------------------------------------------
// Batched tiled GEMM:  C[b] = act( (A[b] @ B[b] + bias) * outScale )  (or += )
//   A element (b,i,k) = A[b*aBatch + i*aRow + k]        (aRow==0 => row bcast)
//   B element (b,k,n) = transB ? Bm[b*bBatch + n*bStride + k]
//                              : Bm[b*bBatch + k*bStride + n]
//   C element (b,i,n) = C[b*cBatch + i*cRow + n]
// Grid: (N/64, rows/128, batch); all dims multiples of tile sizes; K >= TK.
// ---------------------------------------------------------------------------
__global__ __launch_bounds__(256)
void gemm_wmma_f32(const float* __restrict__ A, long long aBatch, long long aRow,
                   const float* __restrict__ Bm, long long bBatch, int bStride, int transB,
                   const float* __restrict__ bias,
                   float* __restrict__ C, long long cBatch, int cRow,
                   int K, int accumulate, int relu, float outScale)
{
    __shared__ float Ash[2][TM * A_STRIDE];          // 2 x 18.0 KB
    __shared__ float Bsh[2][TK * B_STRIDE];          // 2 x  9.0 KB (== 64*BT_STRIDE)

    const int bz       = blockIdx.z;
    const int rowBlock = blockIdx.y * TM;
    const int colBlock = blockIdx.x * TN;
    const float* Ab = A  + (long long)bz * aBatch;
    const float* Bb = Bm + (long long)bz * bBatch;
    float*       Cb = C  + (long long)bz * cBatch;

    const int t    = threadIdx.x;
    const int lane = t & 31;
    const int wave = t >> 5;
    const int wr   = (wave & 3) * 32;   // wave row base within tile
    const int wc   = (wave >> 2) * 32;  // wave col base within tile
    const int lmod = lane & 15;
    const int lhi  = lane >> 4;         // 0 or 1

    // ---- issue async copies for one K-tile into buffer p (6 x b128 / thread)
    auto issue_tile = [&](int p, int k0) {
        const unsigned aBase = lds_off(&Ash[p][0]);
        #pragma unroll
        for (int i = 0; i < 4; ++i) {                // A tile: 128x32 floats
            const int c   = t + 256 * i;
            const int row = c >> 3;
            const int kq  = (c & 7) * 4;
            async_g2l_b128(aBase + 4u * (row * A_STRIDE + kq),
                           Ab + (long long)(rowBlock + row) * aRow + (k0 + kq));
        }
        const unsigned bBase = lds_off(&Bsh[p][0]);
        if (!transB) {                                // B tile: 32x64, [k][n]
            #pragma unroll
            for (int i = 0; i < 2; ++i) {
                const int c  = t + 256 * i;
                const int kk = c >> 4;
                const int nq = (c & 15) * 4;
                async_g2l_b128(bBase + 4u * (kk * B_STRIDE + nq),
                               Bb + (long long)(k0 + kk) * bStride + (colBlock + nq));
            }
        } else {                                      // Bt tile: 64x32, [n][k]
            #pragma unroll
            for (int i = 0; i < 2; ++i) {
                const int c  = t + 256 * i;
                const int n  = c >> 3;
                const int kq = (c & 7) * 4;
                async_g2l_b128(bBase + 4u * (n * BT_STRIDE + kq),
                               Bb + (long long)(colBlock + n) * bStride + (k0 + kq));
            }
        }
    };

    // ---- accumulator init ----
    v8f acc[2][2];
    for (int rt = 0; rt < 2; ++rt)
        for (int ct = 0; ct < 2; ++ct) {
            const int ncol = colBlock + wc + ct * 16 + lmod;
            if (accumulate) {
                #pragma unroll
                for (int r = 0; r < 8; ++r) {
                    const int mrow = rowBlock + wr + rt * 16 + r + 8 * lhi;
                    acc[rt][ct][r] = Cb[(long long)mrow * cRow + ncol];
                }
            } else if (bias) {
                const float bv = bias[ncol];
                #pragma unroll
                for (int r = 0; r < 8; ++r) acc[rt][ct][r] = bv;
            } else {
                #pragma unroll
                for (int r = 0; r < 8; ++r) acc[rt][ct][r] = 0.0f;
            }
        }

    // ---- one K-tile of WMMA work from buffer p ----
    auto compute_tile = [&](int p) {
        #pragma unroll
        for (int kk = 0; kk < TK; kk += 4) {
            v2f afrag[2];
            v2f bfrag[2];
            #pragma unroll
            for (int rt = 0; rt < 2; ++rt) {
                // A layout (16x4 f32): lane L holds M=L%16; VGPR v holds K = v + 2*(L/16)
                const float* ap = &Ash[p][(wr + rt * 16 + lmod) * A_STRIDE + kk + 2 * lhi];
                afrag[rt] = *(const v2f*)ap;       // one ds_load_b64
            }
            if (transB) {
                #pragma unroll
                for (int ct = 0; ct < 2; ++ct) {
                    const float* bp = &Bsh[p][(wc + ct * 16 + lmod) * BT_STRIDE + kk + 2 * lhi];
                    bfrag[ct] = *(const v2f*)bp;   // one ds_load_b64
                }
            } else {
                #pragma unroll
                for (int ct = 0; ct < 2; ++ct) {
                    const int col = wc + ct * 16 + lmod;
                    bfrag[ct].x = Bsh[p][(kk + 2 * lhi + 0) * B_STRIDE + col];
                    bfrag[ct].y = Bsh[p][(kk + 2 * lhi + 1) * B_STRIDE + col];
                }
            }
            #pragma unroll
            for (int rt = 0; rt < 2; ++rt)
                #pragma unroll
                for (int ct = 0; ct < 2; ++ct)
                    acc[rt][ct] = __builtin_amdgcn_wmma_f32_16x16x4_f32(
                        false, afrag[rt], false, bfrag[ct],
                        (short)0, acc[rt][ct], false, false);
        }
    };

    // ---- software-pipelined main loop: last tile peeled (branch-free body) --
    issue_tile(0, 0);
    int p = 0;
    for (int k0 = 0; k0 < K - TK; k0 += TK) {
        issue_tile(p ^ 1, k0 + TK);   // prefetch next tile (ASYNCcnt: 6 cur + 6 next)
        wait_async_6();               // current tile's 6 copies done, next in flight
        __syncthreads();              // all waves' LDS writes visible
        compute_tile(p);
        __syncthreads();              // compute done; buffer reusable
        p ^= 1;
    }
    wait_async_0();                   // final tile complete
    __syncthreads();
    compute_tile(p);

    // ---- epilogue ----
    for (int rt = 0; rt < 2; ++rt)
        for (int ct = 0; ct < 2; ++ct) {
            const int ncol = colBlock + wc + ct * 16 + lmod;
            #pragma unroll
            for (int r = 0; r < 8; ++r) {
                const int mrow = rowBlock + wr + rt * 16 + r + 8 * lhi;
                float v = acc[rt][ct][r] * outScale;
                if (relu) v = fmaxf(v, 0.0f);
                Cb[(long long)mrow * cRow + ncol] = v;
            }
        }
}

// ---------------------------------------------------------------------------
// Row softmax in place (one 256-thread block per row).
// ---------------------------------------------------------------------------
__global__ __launch_bounds__(256)
void softmax_rows(float* __restrict__ x, int cols)
{
    __shared__ float red[256];
    const long long row = blockIdx.x;
    float* p = x + row * (long long)cols;
    const int t = threadIdx.x;

    float mx = -INFINITY;
    for (int c = t; c < cols; c += 256) mx = fmaxf(mx, p[c]);
    red[t] = mx; __syncthreads();
    for (int s = 128; s > 0; s >>= 1) { if (t < s) red[t] = fmaxf(red[t], red[t + s]); __syncthreads(); }
    mx = red[0]; __syncthreads();

    float sum = 0.0f;
    for (int c = t; c < cols; c += 256) { const float e = __expf(p[c] - mx); p[c] = e; sum += e; }
    red[t] = sum; __syncthreads();
    for (int s = 128; s > 0; s >>= 1) { if (t < s) red[t] += red[t + s]; __syncthreads(); }
    const float inv = 1.0f / red[0];
    for (int c = t; c < cols; c += 256) p[c] *= inv;
}

// ---------------------------------------------------------------------------
// prior_absolute = broadcast(anchor) + prior_delta
// ---------------------------------------------------------------------------
__global__ __launch_bounds__(256)
void add_anchor_kernel(const float* __restrict__ delta, const float* __restrict__ anchor,
                       float* __restrict__ out, long long total)
{
    const long long i = (long long)blockIdx.x * 256 + threadIdx.x;
    if (i < total) {
        const int h = (int)(i & (long long)(cH - 1));
        const int b = (int)(i / ((long long)cS * cH));
        out[i] = anchor[b * cH + h] + delta[i];
    }
}

// ---------------------------------------------------------------------------
// Gate head: logit = hid_row . w + b2 ; gate = sigmoid(logit + preBias)
// One wave per row (wave32 shuffle reduction).
// ---------------------------------------------------------------------------
__global__ __launch_bounds__(256)
void gate_head_kernel(const float* __restrict__ hid, const float* __restrict__ w,
                      const float* __restrict__ b2,
                      float* __restrict__ gate_out, float* __restrict__ logit_out,
                      float preBias, int nrows)
{
    const int row = blockIdx.x * 8 + (threadIdx.x >> 5);
    if (row >= nrows) return;
    const int lane = threadIdx.x & 31;
    const float* hp = hid + (long long)row * cGH;
    float s = 0.0f;
    #pragma unroll
    for (int k = 0; k < cGH; k += 32) s += hp[k + lane] * w[k + lane];
    #pragma unroll
    for (int off = 16; off > 0; off >>= 1) s += __shfl_xor(s, off, 32);
    if (lane == 0) {
        const float logit = s + b2[0];
        if (logit_out) logit_out[row] = logit;
        gate_out[row] = 1.0f / (1.0f + __expf(-(logit + preBias)));
    }
}

// ---------------------------------------------------------------------------
// slerp(prior_abs, local_abs, mix) -> candidate_absolute, candidate_delta.
// One 256-thread block per (b,s) row of H=512.
// ---------------------------------------------------------------------------
__global__ __launch_bounds__(256)
void slerp_kernel(const float* __restrict__ pa, const float* __restrict__ la,
                  const float* __restrict__ anchor, const float* __restrict__ mix,
                  float* __restrict__ cand, float* __restrict__ cdelta)
{
    __shared__ float r0[256], r1[256], r2[256];
    const int row = blockIdx.x;
    const int bidx = row / cS;
    const float* a = pa + (long long)row * cH;
    const float* b = la + (long long)row * cH;
    const int t = threadIdx.x;

    float sa = 0.0f, sb = 0.0f, sd = 0.0f;
    for (int h = t; h < cH; h += 256) {
        const float av = a[h], bv = b[h];
        sa += av * av; sb += bv * bv; sd += av * bv;
    }
    r0[t] = sa; r1[t] = sb; r2[t] = sd; __syncthreads();
    for (int s = 128; s > 0; s >>= 1) {
        if (t < s) { r0[t] += r0[t + s]; r1[t] += r1[t + s]; r2[t] += r2[t + s]; }
        __syncthreads();
    }
    const float na = fmaxf(sqrtf(r0[0]), EPS);
    const float nb = fmaxf(sqrtf(r1[0]), EPS);
    float dot = r2[0] / (na * nb);
    dot = fminf(fmaxf(dot, -1.0f + EPS), 1.0f - EPS);
    const float omega = acosf(dot);
    const float sino  = fmaxf(sinf(omega), EPS);
    float tt = mix[row];
    tt = fminf(fmaxf(tt, 0.0f), 1.0f);
    const float s_a = sinf((1.0f - tt) * omega) / sino;
    const float s_b = sinf(tt * omega) / sino;

    for (int h = t; h < cH; h += 256) {
        const float c = s_a * a[h] + s_b * b[h];
        cand[(long long)row * cH + h]   = c;
        cdelta[(long long)row * cH + h] = c - anchor[bidx * cH + h];
    }
}

// ---------------------------------------------------------------------------
extern "C" void kernel_launch(void* const* d_in, const int* in_sizes, int n_in,
                              void* d_out, int out_size, void* d_ws, size_t ws_size,
                              hipStream_t stream)
{
    const float* query    = (const float*)d_in[0];
    const float* anchor   = (const float*)d_in[1];
    const float* memory   = (const float*)d_in[2];
    const float* local_ab = (const float*)d_in[3];
    const float* style    = (const float*)d_in[4];
    const float* Wq = (const float*)d_in[5];  const float* bq = (const float*)d_in[6];
    const float* Wk = (const float*)d_in[7];  const float* bk = (const float*)d_in[8];
    const float* Wv = (const float*)d_in[9];  const float* bv = (const float*)d_in[10];
    const float* Wo = (const float*)d_in[11]; const float* bo = (const float*)d_in[12];
    const float* Wm1 = (const float*)d_in[13]; const float* bm1 = (const float*)d_in[14];
    const float* Wm2 = (const float*)d_in[15]; const float* bm2 = (const float*)d_in[16];
    const float* Wg1 = (const float*)d_in[17]; const float* bg1 = (const float*)d_in[18];
    const float* Wg2 = (const float*)d_in[19]; const float* bg2 = (const float*)d_in[20];

    const long long BS = (long long)cB * cS;   // 16384
    const long long BM = (long long)cB * cM;   // 16384
    const long long SH = (long long)cS * cH;
    const long long MH = (long long)cM * cH;
    const long long SM = (long long)cS * cM;
    const long long SG = (long long)cS * cGH;

    // workspace (floats): q | k | v | ctx | hid  (~168 MB)
    float* q_ws   = (float*)d_ws;
    float* k_ws   = q_ws + BS * cH;
    float* v_ws   = k_ws + BM * cH;
    float* ctx_ws = v_ws + BM * cH;
    float* hid_ws = ctx_ws + BS * cH;

    // output regions (return-order concat, f32)
    float* o_attn  = (float*)d_out;
    float* o_pd    = o_attn + (long long)cB * SM;
    float* o_pa    = o_pd + BS * cH;
    float* o_mix   = o_pa + BS * cH;
    float* o_cand  = o_mix + BS;
    float* o_cd    = o_cand + BS * cH;
    float* o_vlog  = o_cd + BS * cH;
    float* o_vgate = o_vlog + BS;

    const dim3 blk(256);
    const dim3 gProj(cH / TN, (int)(BS / TM), 1);      // (8,128,1)
    const dim3 gLogit(cM / TN, cS / TM, cB);           // (32,16,8)
    const dim3 gCtx(cH / TN, cS / TM, cB);             // (8,16,8)
    const dim3 gGate(cGH / TN, cS / TM, cB);           // (8,16,8)
    const float invSqrtH = 0.044194173824159216f;      // 1/sqrt(512)

    // --- projections: q, k, v -------------------------------------------------
    gemm_wmma_f32<<<gProj, blk, 0, stream>>>(query,  0, cH, Wq, 0, cH, 0, bq,
                                             q_ws, 0, cH, cH, 0, 0, 1.0f);
    gemm_wmma_f32<<<gProj, blk, 0, stream>>>(memory, 0, cH, Wk, 0, cH, 0, bk,
                                             k_ws, 0, cH, cH, 0, 0, 1.0f);
    gemm_wmma_f32<<<gProj, blk, 0, stream>>>(memory, 0, cH, Wv, 0, cH, 0, bv,
                                             v_ws, 0, cH, cH, 0, 0, 1.0f);

    // --- attention logits (B transposed: k is [M,H] row-major) + softmax -----
    gemm_wmma_f32<<<gLogit, blk, 0, stream>>>(q_ws, SH, cH, k_ws, MH, cH, 1, nullptr,
                                              o_attn, SM, cM, cH, 0, 0, invSqrtH);
    softmax_rows<<<dim3((unsigned)BS), blk, 0, stream>>>(o_attn, cM);

    // --- context = attn @ v ---------------------------------------------------
    gemm_wmma_f32<<<gCtx, blk, 0, stream>>>(o_attn, SM, cM, v_ws, MH, cH, 0, nullptr,
                                            ctx_ws, SH, cH, cM, 0, 0, 1.0f);

    // --- prior_delta = (ctx @ Wo + bo) * PRIOR_SCALE; prior_abs = anchor + pd -
    gemm_wmma_f32<<<gProj, blk, 0, stream>>>(ctx_ws, 0, cH, Wo, 0, cH, 0, bo,
                                             o_pd, 0, cH, cH, 0, 0, PRIOR_SCALE);
    add_anchor_kernel<<<dim3((unsigned)(BS * cH / 256)), blk, 0, stream>>>(
        o_pd, anchor, o_pa, BS * cH);

    // --- mix gate hidden = relu([q|pa|la|style] @ Wm1 + bm1), 4 accumulated GEMMs
    gemm_wmma_f32<<<gGate, blk, 0, stream>>>(query, SH, cH, Wm1 + 0 * (long long)cH * cGH,
                                             0, cGH, 0, bm1, hid_ws, SG, cGH, cH, 0, 0, 1.0f);
    gemm_wmma_f32<<<gGate, blk, 0, stream>>>(o_pa, SH, cH, Wm1 + 1 * (long long)cH * cGH,
                                             0, cGH, 0, nullptr, hid_ws, SG, cGH, cH, 1, 0, 1.0f);
    gemm_wmma_f32<<<gGate, blk, 0, stream>>>(local_ab, SH, cH, Wm1 + 2 * (long long)cH * cGH,
                                             0, cGH, 0, nullptr, hid_ws, SG, cGH, cH, 1, 0, 1.0f);
    gemm_wmma_f32<<<gGate, blk, 0, stream>>>(style, SH, cH, Wm1 + 3 * (long long)cH * cGH,
                                             0, cGH, 0, nullptr, hid_ws, SG, cGH, cH, 1, 1, 1.0f);

    // --- mix = sigmoid(hid @ Wm2 + bm2 + MIX_BIAS) ---------------------------
    gate_head_kernel<<<dim3((unsigned)(BS / 8)), blk, 0, stream>>>(
        hid_ws, Wm2, bm2, o_mix, nullptr, MIX_BIAS, (int)BS);

    // --- slerp -> candidate_absolute, candidate_delta ------------------------
    slerp_kernel<<<dim3((unsigned)BS), blk, 0, stream>>>(
        o_pa, local_ab, anchor, o_mix, o_cand, o_cd);

    // --- variation gate hidden = relu([q|cand|anchor|style] @ Wg1 + bg1) -----
    gemm_wmma_f32<<<gGate, blk, 0, stream>>>(query, SH, cH, Wg1 + 0 * (long long)cH * cGH,
                                             0, cGH, 0, bg1, hid_ws, SG, cGH, cH, 0, 0, 1.0f);
    gemm_wmma_f32<<<gGate, blk, 0, stream>>>(o_cand, SH, cH, Wg1 + 1 * (long long)cH * cGH,
                                             0, cGH, 0, nullptr, hid_ws, SG, cGH, cH, 1, 0, 1.0f);
    // anchor_seq: row-broadcast A (aRow = 0, aBatch = H)
    gemm_wmma_f32<<<gGate, blk, 0, stream>>>(anchor, cH, 0, Wg1 + 2 * (long long)cH * cGH,
                                             0, cGH, 0, nullptr, hid_ws, SG, cGH, cH, 1, 0, 1.0f);
    gemm_wmma_f32<<<gGate, blk, 0, stream>>>(style, SH, cH, Wg1 + 3 * (long long)cH * cGH,
                                             0, cGH, 0, nullptr, hid_ws, SG, cGH, cH, 1, 1, 1.0f);

    // --- variation_logit / variation_gate ------------------------------------
    gate_head_kernel<<<dim3((unsigned)(BS / 8)), blk, 0, stream>>>(
        hid_ws, Wg2, bg2, o_vgate, o_vlog, VARIATION_BIAS, (int)BS);
}